// CARLayer_56298431316390
// MI455X (gfx1250) — compile-verified
//
#include <hip/hip_runtime.h>
#include <math.h>

typedef __attribute__((ext_vector_type(2))) float v2f;
typedef __attribute__((ext_vector_type(8))) float v8f;

#define DD   512
#define SS   4096
#define NB   16
#define NBR  4

#define CH   256          // seq rows per block (split-K chunk)
#define TR   16           // rows per LDS tile (= #waves per block)
#define NCH  (SS / CH)    // 16 chunks per (branch,batch)
#define NT   (CH / TR)    // 16 tiles per block

// ---------------------------------------------------------------- CDNA5 async helpers
// One instruction per wave: each lane copies 16 bytes global -> LDS. Tracked by ASYNCcnt.
__device__ __forceinline__ void async_ld_b128(const void* gptr, void* lptr) {
    // Generic LDS pointer: ADDR[31:0] is the LDS byte offset (aperture in high bits).
    uint32_t loff = (uint32_t)(uintptr_t)lptr;
    asm volatile("global_load_async_to_lds_b128 %0, %1, off"
                 :: "v"(loff), "v"(gptr) : "memory");
}

template <int N>
__device__ __forceinline__ void wait_async() {
#if __has_builtin(__builtin_amdgcn_s_wait_asynccnt)
    __builtin_amdgcn_s_wait_asynccnt(N);
#else
    asm volatile("s_wait_asynccnt %0" :: "i"(N) : "memory");
#endif
}

// ------------------------------------------------- v[br,b,:] = K @ c[br,b,:] + bias
// One wave32 per (branch, 16-row d-tile). V_WMMA_F32_16X16X4_F32, K-loop over e.
__global__ __launch_bounds__(32) void compute_v_kernel(
    const float* __restrict__ Kmat,   // (D,D) row-major [d][e]
    const float* __restrict__ bias,   // (D)
    const float* __restrict__ c0, const float* __restrict__ c1,
    const float* __restrict__ c2, const float* __restrict__ c3,
    float* __restrict__ vws)          // (NBR, NB, D)
{
    const int blk  = blockIdx.x;          // 0..127
    const int br   = blk >> 5;            // branch
    const int dt   = blk & 31;            // d-tile (16 rows)
    const float* c = (br == 0) ? c0 : (br == 1) ? c1 : (br == 2) ? c2 : c3;

    const int lane = threadIdx.x;         // 0..31 (wave32)
    const int half = lane >> 4;           // 0 or 1
    const int l16  = lane & 15;
    const int arow = dt * 16 + l16;       // d row of A-fragment

    v8f acc = {};
    for (int e0 = 0; e0 < DD; e0 += 4) {
        const int kb = e0 + 2 * half;
        v2f a = *reinterpret_cast<const v2f*>(Kmat + (size_t)arow * DD + kb);
        v2f b = *reinterpret_cast<const v2f*>(c    + (size_t)l16  * DD + kb);
        acc = __builtin_amdgcn_wmma_f32_16x16x4_f32(
            false, a, false, b, (short)0, acc, false, false);
    }

    const int mbase = half * 8;
    #pragma unroll
    for (int r = 0; r < 8; ++r) {
        const int d = dt * 16 + mbase + r;
        vws[((size_t)br * NB + l16) * DD + d] = acc[r] + bias[d];
    }
}

// ------------------------------------------------- fused flash pass (single HBM read of sent)
// Block = (chunk, batch, branch); 512 threads = 16 waves. Per 16-row tile:
//   async global->LDS (double buffered), wave w dots row w with v, then all
//   threads do an online-softmax update of (m, l, acc[d=tid]).
__global__ __launch_bounds__(512) void flash_kernel(
    const float* __restrict__ s0, const float* __restrict__ s1,
    const float* __restrict__ s2, const float* __restrict__ s3,
    const float* __restrict__ vws,     // (NBR, NB, D)
    float* __restrict__ pacc,          // (NBR*NB, NCH, D)
    float* __restrict__ pml)           // (NBR*NB, NCH, 2) : {m, l}
{
    const int br    = blockIdx.z;
    const int bb    = blockIdx.y;
    const int chunk = blockIdx.x;
    const float* sent = (br == 0) ? s0 : (br == 1) ? s1 : (br == 2) ? s2 : s3;

    const int t    = threadIdx.x;        // 0..511 ; owns d = t in accumulate phase
    const int wv   = t >> 5;             // wave 0..15 ; owns row wv in dot phase
    const int lane = t & 31;
    const int gb   = br * NB + bb;
    const float* vrow = vws + (size_t)gb * DD;

    __shared__ __align__(16) float buf[2][TR * DD];   // 2 x 32KB tiles
    __shared__ float scl[TR];

    const char* gbase =
        (const char*)(sent + ((size_t)bb * SS + (size_t)chunk * CH) * DD);

    // prologue: tile 0 -> buf[0]  (4 async ops per wave, lane-parallel 16B each)
    {
        char* l0 = (char*)&buf[0][0];
        #pragma unroll
        for (int k = 0; k < 4; ++k)
            async_ld_b128(gbase + k * 8192 + t * 16, l0 + k * 8192 + t * 16);
    }

    float m = -INFINITY, l = 0.0f, acc = 0.0f;

    for (int tile = 0; tile < NT; ++tile) {
        const int cur = tile & 1;

        if (tile + 1 < NT) {                         // prefetch next tile
            char* ln = (char*)&buf[cur ^ 1][0];
            const char* gn = gbase + (size_t)(tile + 1) * TR * DD * 4;
            #pragma unroll
            for (int k = 0; k < 4; ++k)
                async_ld_b128(gn + k * 8192 + t * 16, ln + k * 8192 + t * 16);
            wait_async<4>();                         // current tile done; next in flight
        } else {
            wait_async<0>();
        }
        __syncthreads();                             // make LDS tile visible to all waves

        // --- dot phase: wave wv computes score of row wv from LDS
        float d0 = 0.0f;
        const float* row = &buf[cur][wv * DD];
        #pragma unroll
        for (int i = 0; i < 4; ++i) {
            float4 x  = *reinterpret_cast<const float4*>(row  + i * 128 + lane * 4);
            float4 w4 = *reinterpret_cast<const float4*>(vrow + i * 128 + lane * 4);
            d0 += x.x * w4.x + x.y * w4.y + x.z * w4.z + x.w * w4.w;
        }
        #pragma unroll
        for (int off = 16; off > 0; off >>= 1) d0 += __shfl_xor(d0, off, 32);
        if (lane == 0) scl[wv] = d0;
        __syncthreads();                             // scores visible

        // --- online softmax accumulate: thread owns column d = t
        float nm = m;
        #pragma unroll
        for (int r = 0; r < TR; ++r) nm = fmaxf(nm, scl[r]);
        const float scale = __expf(m - nm);          // first tile: exp(-inf)=0
        acc *= scale;
        l   *= scale;
        #pragma unroll
        for (int r = 0; r < TR; ++r) {
            const float wgt = __expf(scl[r] - nm);
            l += wgt;
            acc = fmaf(buf[cur][r * DD + t], wgt, acc);
        }
        m = nm;
        __syncthreads();                             // done reading buf[cur] before reuse
    }

    pacc[((size_t)gb * NCH + chunk) * DD + t] = acc;
    if (t == 0) {
        pml[((size_t)gb * NCH + chunk) * 2 + 0] = m;
        pml[((size_t)gb * NCH + chunk) * 2 + 1] = l;
    }
}

// ------------------------------------------------- split-K combine: out = sum_i acc_i*e^{m_i-M} / L
__global__ __launch_bounds__(512) void combine_kernel(
    const float* __restrict__ pacc, const float* __restrict__ pml,
    float* __restrict__ out)
{
    const int gb = blockIdx.x;    // 0..63 (branch*16 + batch)
    const int t  = threadIdx.x;   // d

    float M = -INFINITY;
    #pragma unroll
    for (int i = 0; i < NCH; ++i)
        M = fmaxf(M, pml[((size_t)gb * NCH + i) * 2]);

    float L = 0.0f, acc = 0.0f;
    #pragma unroll
    for (int i = 0; i < NCH; ++i) {
        const float sc = __expf(pml[((size_t)gb * NCH + i) * 2] - M);
        L += pml[((size_t)gb * NCH + i) * 2 + 1] * sc;
        acc = fmaf(pacc[((size_t)gb * NCH + i) * DD + t], sc, acc);
    }
    out[(size_t)gb * DD + t] = acc / L;
}

// ---------------------------------------------------------------- launch
extern "C" void kernel_launch(void* const* d_in, const int* in_sizes, int n_in,
                              void* d_out, int out_size, void* d_ws, size_t ws_size,
                              hipStream_t stream) {
    // setup_inputs order: sent1, c1, sent2, c2, sent3, c3, sent4, c4, kernel, bias
    const float* sent[4] = { (const float*)d_in[0], (const float*)d_in[2],
                             (const float*)d_in[4], (const float*)d_in[6] };
    const float* cin[4]  = { (const float*)d_in[1], (const float*)d_in[3],
                             (const float*)d_in[5], (const float*)d_in[7] };
    const float* Kmat = (const float*)d_in[8];
    const float* bias = (const float*)d_in[9];
    float* out = (float*)d_out;

    // workspace (floats): v (4*16*512) | pacc (64*16*512) | pml (64*16*2)  ~2.3 MB
    float* vws  = (float*)d_ws;
    float* pacc = vws + NBR * NB * DD;
    float* pml  = pacc + (size_t)NBR * NB * NCH * DD;

    compute_v_kernel<<<NBR * (DD / 16), 32, 0, stream>>>(
        Kmat, bias, cin[0], cin[1], cin[2], cin[3], vws);

    dim3 gf(NCH, NB, NBR);   // 16 x 16 x 4 = 1024 blocks
    flash_kernel<<<gf, 512, 0, stream>>>(
        sent[0], sent[1], sent[2], sent[3], vws, pacc, pml);

    combine_kernel<<<NBR * NB, 512, 0, stream>>>(pacc, pml, out);
}